// PointHead_5488968204893
// MI455X (gfx1250) — compile-verified
//
#include <hip/hip_runtime.h>
#include <math.h>

// ---------------------------------------------------------------------------
// PointHead (PointRend-style) for MI455X / gfx1250.
//
// Key simplifications proven from the reference:
//  * input x is only used for its shape (N = 128) -> never read.
//  * "coverage" coords trunc() to exactly 0 for every possible top-k index,
//    so the Sobel/softmax/top-k edge branch is dead code -> skipped.
//  * point-sampling is a sparse gather; we never scan res2 (268MB) or out.
//
// MLP runs on V_WMMA_F32_16X16X4_F32 (f32 keeps reference numerics; the
// whole workload is ~0.8 GFLOP / ~10MB => latency bound, so precision wins
// over fp16 throughput). GEMM v2: one workgroup per 16-row M strip, A strip
// staged once in LDS (coalesced, shared by all 8 waves = all n-tiles),
// conflict-free LDS pitch of 516 dwords, A fragment reused across NTW
// n-tiles per k-step, W panel streamed from L2 with global_prefetch ahead.
// ---------------------------------------------------------------------------

typedef float v2f __attribute__((ext_vector_type(2)));
typedef float v8f __attribute__((ext_vector_type(8)));

#define BATCH   8
#define IN_C    512
#define NCLS    2
#define HF      128
#define NPTS    128      // N
#define NIMP    96
#define NCOV    16
#define NRAND   16
#define NOVER   384      // 3*N candidate points
#define K1      514      // IN_C + NCLS
#define K1P     516      // K1 padded to multiple of 4 (WMMA K-step)
#define H1DIM   512
#define H2DIM   256
#define MROWS   (BATCH * NPTS)   // 1024
#define LDSPITCH 516     // dword pitch: 516 % 64 == 4 -> conflict-free frags

// Bilinear sampling setup matching grid_sample(align_corners=False, zero pad)
__device__ __forceinline__ void corner_setup(float px, float py,
                                             int lin4[4], float w4[4]) {
  float gx = px * (float)HF - 0.5f;
  float gy = py * (float)HF - 0.5f;
  float x0 = floorf(gx), y0 = floorf(gy);
  float fx = gx - x0,    fy = gy - y0;
  float xs[2] = {x0, x0 + 1.0f};
  float ys[2] = {y0, y0 + 1.0f};
  float wx[2] = {1.0f - fx, fx};
  float wy[2] = {1.0f - fy, fy};
  int q = 0;
  for (int j = 0; j < 2; ++j) {
    bool vy = (ys[j] >= 0.0f) && (ys[j] <= (float)(HF - 1));
    int  yi = (int)fminf(fmaxf(ys[j], 0.0f), (float)(HF - 1));
    for (int i = 0; i < 2; ++i) {
      bool vx = (xs[i] >= 0.0f) && (xs[i] <= (float)(HF - 1));
      int  xi = (int)fminf(fmaxf(xs[i], 0.0f), (float)(HF - 1));
      w4[q]   = (vx && vy) ? wx[i] * wy[j] : 0.0f;
      lin4[q] = yi * HF + xi;
      ++q;
    }
  }
}

// ---------------------------------------------------------------------------
// Kernel 1: zero-pad w1 rows 514..515 (so GEMM1 K is a multiple of 4).
// ---------------------------------------------------------------------------
__global__ void pad_w1_kernel(const float* __restrict__ w1,
                              float* __restrict__ w1p) {
  int i = blockIdx.x * blockDim.x + threadIdx.x;
  if (i >= K1P * H1DIM) return;
  w1p[i] = (i < K1 * H1DIM) ? w1[i] : 0.0f;
}

// ---------------------------------------------------------------------------
// Kernel 2: sampling points. One block per batch, 384 threads (one per
// over_gen candidate). Stable top-96 via O(n^2) ranking (lax.top_k order).
// ---------------------------------------------------------------------------
__global__ void points_kernel(const float* __restrict__ out,
                              const float* __restrict__ over_gen,
                              const float* __restrict__ rand_point,
                              float* __restrict__ pts_ws,
                              float* __restrict__ pts_out) {
  __shared__ float s_unc[NOVER];
  int b = blockIdx.x, t = threadIdx.x;
  const float* outb = out + (size_t)b * NCLS * HF * HF;

  float px = over_gen[(b * NOVER + t) * 2 + 0];
  float py = over_gen[(b * NOVER + t) * 2 + 1];
  int lin4[4]; float w4[4];
  corner_setup(px, py, lin4, w4);

  float og0 = 0.0f, og1 = 0.0f;   // bilinear of per-pixel max / min channel
  for (int q = 0; q < 4; ++q) {
    float a = outb[lin4[q]];
    float c = outb[HF * HF + lin4[q]];
    og0 += w4[q] * fmaxf(a, c);
    og1 += w4[q] * fminf(a, c);
  }
  s_unc[t] = og1 - og0;           // unc = -(max - min)
  __syncthreads();

  float mine = s_unc[t];
  int rank = 0;
  for (int j = 0; j < NOVER; ++j) {
    float v = s_unc[j];
    rank += (int)((v > mine) || (v == mine && j < t));
  }
  if (rank < NIMP) {
    int idx = (b * NPTS + rank) * 2;
    pts_ws[idx + 0] = px;  pts_ws[idx + 1] = py;
    pts_out[idx + 0] = px; pts_out[idx + 1] = py;
  }
  if (t < NCOV) {                 // coverage: provably (0,0)
    int idx = (b * NPTS + NIMP + t) * 2;
    pts_ws[idx + 0] = 0.0f;  pts_ws[idx + 1] = 0.0f;
    pts_out[idx + 0] = 0.0f; pts_out[idx + 1] = 0.0f;
  }
  if (t < NRAND) {
    int idx = (b * NPTS + NIMP + NCOV + t) * 2;
    float rx = rand_point[(b * NRAND + t) * 2 + 0];
    float ry = rand_point[(b * NRAND + t) * 2 + 1];
    pts_ws[idx + 0] = rx;  pts_ws[idx + 1] = ry;
    pts_out[idx + 0] = rx; pts_out[idx + 1] = ry;
  }
}

// ---------------------------------------------------------------------------
// Kernel 3: sparse feature gather. One block per (b,n) point; 128 threads
// stride over 516 channels (2 coarse + 512 fine + 2 zero pad).
// ---------------------------------------------------------------------------
__global__ void gather_kernel(const float* __restrict__ out,
                              const float* __restrict__ res2,
                              const float* __restrict__ pts,
                              float* __restrict__ feat,
                              float* __restrict__ coarse) {
  int bn = blockIdx.x;
  int b = bn / NPTS, n = bn % NPTS;
  float px = pts[bn * 2 + 0], py = pts[bn * 2 + 1];
  int lin4[4]; float w4[4];
  corner_setup(px, py, lin4, w4);

  for (int c = threadIdx.x; c < K1P; c += blockDim.x) {
    float v = 0.0f;
    if (c < NCLS) {
      const float* p = out + ((size_t)b * NCLS + c) * HF * HF;
      v = w4[0]*p[lin4[0]] + w4[1]*p[lin4[1]] + w4[2]*p[lin4[2]] + w4[3]*p[lin4[3]];
      coarse[b * NCLS * NPTS + c * NPTS + n] = v;
    } else if (c < K1) {
      const float* p = res2 + ((size_t)b * IN_C + (c - NCLS)) * HF * HF;
      v = w4[0]*p[lin4[0]] + w4[1]*p[lin4[1]] + w4[2]*p[lin4[2]] + w4[3]*p[lin4[3]];
    }
    feat[(size_t)bn * K1P + c] = v;
  }
}

// ---------------------------------------------------------------------------
// Kernel 4: f32 WMMA GEMM with LDS-staged A strip.
//   Out[M,N] = A[M,K] @ W[K,N] + bias, one workgroup per 16-row M strip.
//   * A strip [16 x K] staged once into LDS (coalesced), pitch 516 dwords
//     (516 % 64banks == 4 -> half-wave fragment reads hit all 64 banks
//     exactly once, conflict-free for both K=516 and K=512).
//   * 8 waves cover all N/16 n-tiles (NTW tiles each); each A fragment read
//     from LDS feeds NTW WMMAs.
//   * W panel streamed from global (coalesced, L2-resident) with
//     global_prefetch a few k-rows ahead.
// VGPR layouts per CDNA5 ISA 7.12.2:
//   A 16x4 : lanes 0-15 hold K=k0,k0+1; lanes 16-31 hold K=k0+2,k0+3
//   B 4x16 : mirrored; C/D: VGPR j = rows j / j+8 split at lane 16.
// Control flow is wave-uniform so EXEC stays all-1s (WMMA requirement).
// ---------------------------------------------------------------------------
template <int NTW>
__global__ void gemm_wmma_lds(const float* __restrict__ A,
                              const float* __restrict__ W,
                              const float* __restrict__ bias,
                              float* __restrict__ Out,
                              int K, int N) {
  __shared__ float As[16 * LDSPITCH];            // 33 KB of the 320KB WGP LDS
  int mt  = blockIdx.x;
  int tid = threadIdx.x;

  // cooperative, coalesced stage of the 16 x K A strip
  const float* Ab = A + (size_t)mt * 16 * K;
  for (int r = 0; r < 16; ++r)
    for (int c = tid; c < K; c += 256)
      As[r * LDSPITCH + c] = Ab[(size_t)r * K + c];
  __syncthreads();

  int wave = tid >> 5;
  int lane = tid & 31;
  int hl = lane & 15;                  // position within half-wave
  int hi = lane >> 4;                  // 0: K pair {0,1}, 1: K pair {2,3}

  v8f acc[NTW];
#pragma unroll
  for (int q = 0; q < NTW; ++q)
    acc[q] = (v8f){0.f, 0.f, 0.f, 0.f, 0.f, 0.f, 0.f, 0.f};

  const float* Alds  = As + hl * LDSPITCH + 2 * hi;
  const float* Wbase = W + (size_t)(2 * hi) * N + (wave * NTW) * 16 + hl;

  for (int k0 = 0; k0 < K; k0 += 4) {
    v2f a;
    a.x = Alds[k0];                    // ds_load, conflict-free
    a.y = Alds[k0 + 1];
    if (k0 + 8 < K)                    // global_prefetch_b8 of W panel ahead
      __builtin_prefetch(Wbase + (size_t)(k0 + 8) * N, 0, 3);
#pragma unroll
    for (int q = 0; q < NTW; ++q) {    // reuse A fragment across NTW n-tiles
      const float* Wc = Wbase + q * 16;
      v2f bf;
      bf.x = Wc[(size_t)k0 * N];
      bf.y = Wc[(size_t)k0 * N + N];
      acc[q] = __builtin_amdgcn_wmma_f32_16x16x4_f32(
          /*neg_a=*/false, a, /*neg_b=*/false, bf,
          /*c_mod=*/(short)0, acc[q], /*reuse_a=*/false, /*reuse_b=*/false);
    }
  }

#pragma unroll
  for (int q = 0; q < NTW; ++q) {
    int col = (wave * NTW + q) * 16 + hl;
    float bv = bias[col];
#pragma unroll
    for (int j = 0; j < 8; ++j) {
      int m = mt * 16 + j + 8 * hi;
      Out[(size_t)m * N + col] = acc[q][j] + bv;
    }
  }
}

// ---------------------------------------------------------------------------
// Kernel 5: in-place LayerNorm (two-pass, matching reference) + exact GELU.
// One 256-thread block per row; L in {512, 256}.
// ---------------------------------------------------------------------------
__global__ void ln_gelu_kernel(float* __restrict__ X,
                               const float* __restrict__ g,
                               const float* __restrict__ be,
                               int L) {
  __shared__ float red[256];
  int row = blockIdx.x, t = threadIdx.x;
  float* x = X + (size_t)row * L;
  int per = L >> 8;                     // 2 or 1
  float local[2];

  float s = 0.0f;
  for (int i = 0; i < per; ++i) { local[i] = x[t + i * 256]; s += local[i]; }
  red[t] = s; __syncthreads();
  for (int off = 128; off > 0; off >>= 1) {
    if (t < off) red[t] += red[t + off];
    __syncthreads();
  }
  float mean = red[0] / (float)L;
  __syncthreads();

  float s2 = 0.0f;
  for (int i = 0; i < per; ++i) { float d = local[i] - mean; s2 += d * d; }
  red[t] = s2; __syncthreads();
  for (int off = 128; off > 0; off >>= 1) {
    if (t < off) red[t] += red[t + off];
    __syncthreads();
  }
  float rstd = rsqrtf(red[0] / (float)L + 1e-5f);

  for (int i = 0; i < per; ++i) {
    int c = t + i * 256;
    float y = (local[i] - mean) * rstd * g[c] + be[c];
    x[c] = 0.5f * y * (1.0f + erff(y * 0.70710678118654752f)); // exact GELU
  }
}

// ---------------------------------------------------------------------------
// Kernel 6: final projection (N=2, too narrow for WMMA) fused with dropout
// mask and coarse residual. One block per batch; thread = (class, point).
// ---------------------------------------------------------------------------
__global__ void final_kernel(const float* __restrict__ h2,
                             const float* __restrict__ w3,
                             const float* __restrict__ b3,
                             const float* __restrict__ mask,
                             const float* __restrict__ coarse,
                             float* __restrict__ rend) {
  int b = blockIdx.x, t = threadIdx.x;
  int c = t >> 7, n = t & 127;
  const float* hrow = h2 + (size_t)(b * NPTS + n) * H2DIM;
  float s = b3[c];
  for (int k = 0; k < H2DIM; ++k) s += hrow[k] * w3[k * NCLS + c];
  int idx = b * NCLS * NPTS + c * NPTS + n;
  rend[idx] = s * mask[idx] + coarse[idx];
}

// ---------------------------------------------------------------------------
extern "C" void kernel_launch(void* const* d_in, const int* in_sizes, int n_in,
                              void* d_out, int out_size, void* d_ws, size_t ws_size,
                              hipStream_t stream) {
  (void)in_sizes; (void)n_in; (void)out_size; (void)ws_size;
  // d_in[0] = x : used only for its shape in the reference -> never read.
  const float* res2       = (const float*)d_in[1];
  const float* out        = (const float*)d_in[2];
  const float* over_gen   = (const float*)d_in[3];
  const float* rand_point = (const float*)d_in[4];
  const float* mask       = (const float*)d_in[5];
  const float* w1  = (const float*)d_in[6];
  const float* b1  = (const float*)d_in[7];
  const float* g1  = (const float*)d_in[8];
  const float* be1 = (const float*)d_in[9];
  const float* w2  = (const float*)d_in[10];
  const float* b2  = (const float*)d_in[11];
  const float* g2  = (const float*)d_in[12];
  const float* be2 = (const float*)d_in[13];
  const float* w3  = (const float*)d_in[14];
  const float* b3  = (const float*)d_in[15];

  float* rend    = (float*)d_out;                     // [B,C,N]   2048 floats
  float* pts_out = rend + BATCH * NCLS * NPTS;        // [B,N,2]   2048 floats

  // workspace carve-up (~6.04 MB of floats)
  float* ws     = (float*)d_ws;
  float* pts_ws = ws;  ws += BATCH * NPTS * 2;            // 2048
  float* coarse = ws;  ws += BATCH * NCLS * NPTS;         // 2048
  float* w1p    = ws;  ws += (size_t)K1P * H1DIM;         // 264192
  float* feat   = ws;  ws += (size_t)MROWS * K1P;         // 528384
  float* h1     = ws;  ws += (size_t)MROWS * H1DIM;       // 524288
  float* h2     = ws;  ws += (size_t)MROWS * H2DIM;       // 262144

  pad_w1_kernel<<<(K1P * H1DIM + 255) / 256, 256, 0, stream>>>(w1, w1p);
  points_kernel<<<BATCH, NOVER, 0, stream>>>(out, over_gen, rand_point,
                                             pts_ws, pts_out);
  gather_kernel<<<BATCH * NPTS, 128, 0, stream>>>(out, res2, pts_ws,
                                                  feat, coarse);
  // GEMM1: [1024 x 516] @ [516 x 512]; 64 blocks, 8 waves x 4 n-tiles each
  gemm_wmma_lds<4><<<MROWS / 16, 256, 0, stream>>>(feat, w1p, b1, h1,
                                                   K1P, H1DIM);
  ln_gelu_kernel<<<MROWS, 256, 0, stream>>>(h1, g1, be1, H1DIM);
  // GEMM2: [1024 x 512] @ [512 x 256]; 64 blocks, 8 waves x 2 n-tiles each
  gemm_wmma_lds<2><<<MROWS / 16, 256, 0, stream>>>(h1, w2, b2, h2,
                                                   H1DIM, H2DIM);
  ln_gelu_kernel<<<MROWS, 256, 0, stream>>>(h2, g2, be2, H2DIM);
  final_kernel<<<BATCH, 256, 0, stream>>>(h2, w3, b3, mask, coarse, rend);
}